// NSFPProcessor_64596308132008
// MI455X (gfx1250) — compile-verified
//
#include <hip/hip_runtime.h>
#include <hip/hip_bf16.h>

typedef __attribute__((ext_vector_type(2))) float v2f;
typedef __attribute__((ext_vector_type(8))) float v8f;

#define TILES_PER_WAVE   2
#define WAVES_PER_BLOCK  8
#define ROWS_PER_BLOCK   (TILES_PER_WAVE * WAVES_PER_BLOCK * 16)   // 256

// Chamfer direction kernel. One wave owns TILES_PER_WAVE row-tiles of 16 query
// points and sweeps all 16-point column tiles of the reference cloud.
//
// Trick: V_WMMA_F32_16X16X4_F32 with
//   A row  m = (-2*q0, -2*q1, -2*q2, 1)        (16x4, K=3 slot carries the "1")
//   B col  n = ( r0,    r1,    r2,  |r|^2 )    (4x16, K=3 slot carries |r|^2)
// gives D[m][n] = |r_n|^2 - 2*q_m.r_n in a single matrix op; adding the row
// constant |q_m|^2 after the min gives the squared distance. The inner loop is
// branchless and has zero LDS traffic.
__global__ __launch_bounds__(256)
void chamfer_dir_kernel(const float* __restrict__ pc1,
                        const float* __restrict__ pc2,
                        int np, float invP,
                        float* __restrict__ out)
{
    // blockIdx.y selects direction: 0 = pc1->pc2 (cham_x), 1 = pc2->pc1 (cham_y)
    const float* __restrict__ Q = (blockIdx.y == 0) ? pc1 : pc2;
    const float* __restrict__ R = (blockIdx.y == 0) ? pc2 : pc1;

    const int lane = threadIdx.x & 31;
    const int wave = threadIdx.x >> 5;
    const int lp   = lane & 15;   // point index within 16-wide tile
    const bool hiH = (lane >> 4) != 0; // false: K={0,1}, true: K={2,3}

    const int rowBase = blockIdx.x * ROWS_PER_BLOCK + wave * (TILES_PER_WAVE * 16);

    v2f   a[TILES_PER_WAVE];        // A tiles (16x4 f32), resident all loop long
    float q2v[TILES_PER_WAVE][8];   // |q|^2 arranged to match C/D row layout
    v8f   tmin[TILES_PER_WAVE];     // running min of (|r|^2 - 2*q.r) per C/D slot

    #pragma unroll
    for (int rt = 0; rt < TILES_PER_WAVE; ++rt) {
        const int qi = rowBase + rt * 16 + lp;
        const float* qp = Q + qi * 3;
        const float c0 = qp[0], c1 = qp[1], c2 = qp[2];
        // A layout (ISA 7.12.2): lanes 0-15 VGPR i = K=i, lanes 16-31 = K=i+2
        a[rt].x = hiH ? (-2.0f * c2) : (-2.0f * c0);
        a[rt].y = hiH ? 1.0f         : (-2.0f * c1);
        const float q2p = c0 * c0 + c1 * c1 + c2 * c2;   // full |q|^2, no shuffle
        // C/D layout: VGPR i on half 'hi' holds row M = i + 8*hi
        #pragma unroll
        for (int i = 0; i < 8; ++i)
            q2v[rt][i] = __shfl(q2p, i + (hiH ? 8 : 0), 32);
        #pragma unroll
        for (int i = 0; i < 8; ++i)
            tmin[rt][i] = 3.0e38f;
    }

    const int ntiles = np >> 4;
    for (int ct = 0; ct < ntiles; ++ct) {
        const int ri = (ct << 4) + lp;
        const float* rp = R + ri * 3;
        // Unconditional 12B load per lane (merges to global_load_b96), branchless.
        const float c0 = rp[0], c1 = rp[1], c2 = rp[2];
        // stream ahead (global_prefetch_b8); speculative, OOB is dropped
        __builtin_prefetch(rp + 768, 0, 1);
        const float s = c0 * c0 + c1 * c1 + c2 * c2;     // |r|^2
        v2f b;
        // B layout: VGPR i holds row K=i on lanes 0-15, K=i+2 on lanes 16-31
        b.x = hiH ? c2 : c0;
        b.y = hiH ? s  : c1;

        #pragma unroll
        for (int rt = 0; rt < TILES_PER_WAVE; ++rt) {
            v8f c = {};
            // D = A(16x4) * B(4x16) + 0  ->  |r|^2 - 2*q.r per element
            c = __builtin_amdgcn_wmma_f32_16x16x4_f32(
                    false, a[rt], false, b, (short)0, c, false, false);
            #pragma unroll
            for (int i = 0; i < 8; ++i)
                tmin[rt][i] = fminf(tmin[rt][i], c[i]);
        }
    }

    // Cross-lane min within each 16-lane half (same M rows), then threshold+sum.
    float waveSum = 0.0f;
    #pragma unroll
    for (int rt = 0; rt < TILES_PER_WAVE; ++rt) {
        #pragma unroll
        for (int i = 0; i < 8; ++i) {
            float m = tmin[rt][i];
            #pragma unroll
            for (int off = 1; off < 16; off <<= 1)
                m = fminf(m, __shfl_xor(m, off, 32));
            float d = q2v[rt][i] + m;                    // add row constant |q|^2
            waveSum += (d < 2.0f) ? d : 0.0f;            // threshold mask
        }
    }
    // half 0 carries rows M=0..7 (+16*rt), half 1 carries rows M=8..15 (+16*rt)
    float otherHalf = __shfl_xor(waveSum, 16, 32);
    if (lane == 0) {
        atomicAdd(out, (waveSum + otherHalf) * invP);
    }
}

extern "C" void kernel_launch(void* const* d_in, const int* in_sizes, int n_in,
                              void* d_out, int out_size, void* d_ws, size_t ws_size,
                              hipStream_t stream) {
    const float* pc1 = (const float*)d_in[0];
    const float* pc2 = (const float*)d_in[1];
    float* out = (float*)d_out;

    const int np = in_sizes[0] / 3;          // 16384 points per cloud
    const float invP = 1.0f / (float)np;     // point_reduction='mean', N=1

    hipMemsetAsync(out, 0, sizeof(float), stream);

    dim3 grid(np / ROWS_PER_BLOCK, 2);       // y: two chamfer directions
    chamfer_dir_kernel<<<grid, 256, 0, stream>>>(pc1, pc2, np, invP, out);
}